// ExponentialMovingAverage_8572754723403
// MI455X (gfx1250) — compile-verified
//
#include <hip/hip_runtime.h>

typedef __attribute__((ext_vector_type(4))) float f32x4;

// Problem shape (fixed by the reference).
constexpr int B = 32;
constexpr int T = 8192;
constexpr int C = 128;

// Chunked-scan decomposition.
constexpr int K = 128;        // chunks along T
constexpr int L = T / K;      // 64 timesteps per chunk
constexpr int CV = 4;         // channels per thread (one b128 per lane)
constexpr int CG = C / CV;    // 32 channel-groups -> one wave covers a full 512B row
constexpr int TS = C / 4;     // time stride in f32x4 units

// EMA affine map over a chunk: a_out = d^L * a_in + s  (s = scan from zero state).

__device__ __forceinline__ f32x4 clamp01(f32x4 v) {
    v.x = fminf(fmaxf(v.x, 0.f), 1.f);
    v.y = fminf(fmaxf(v.y, 0.f), 1.f);
    v.z = fminf(fmaxf(v.z, 0.f), 1.f);
    v.w = fminf(fmaxf(v.w, 0.f), 1.f);
    return v;
}

// ---------------------------------------------------------------------------
// Kernel 1: per-(b, chunk, channel-quad) partial offset s (entry state = 0).
// Regular-temporal loads: leave the input resident in L2 for kernel 3.
// ws layout: float carries[K][B][C]
// ---------------------------------------------------------------------------
__global__ __launch_bounds__(256) void ema_partials(
    const float* __restrict__ inp,      // [B,T,C]
    const float* __restrict__ weights,  // [C]
    float* __restrict__ carries)        // [K,B,C]
{
    const int tid = blockIdx.x * blockDim.x + threadIdx.x;  // 0 .. B*K*CG-1
    const int cg  = tid & (CG - 1);                         // channel group
    const int bk  = tid >> 5;                               // (b,k) index
    const int k   = bk & (K - 1);
    const int b   = bk >> 7;                                // bk / K
    const int c0  = cg * CV;

    const f32x4 w4 = clamp01(*(const f32x4*)(weights + c0));
    const f32x4 d4 = {1.f - w4.x, 1.f - w4.y, 1.f - w4.z, 1.f - w4.w};

    const f32x4* __restrict__ x4 =
        (const f32x4*)(inp + ((size_t)b * T + (size_t)k * L) * C + c0);

    f32x4 s = {0.f, 0.f, 0.f, 0.f};

    // gfx1250 global_prefetch_b8 (SYS scope -> pull into GL2 ahead of stream).
    __builtin_prefetch(x4 + 16 * TS, 0, 0);
    __builtin_prefetch(x4 + 32 * TS, 0, 0);

#pragma unroll 16
    for (int i = 0; i < L; ++i) {
        if ((i & 15) == 0) __builtin_prefetch(x4 + (i + 48) * TS, 0, 0);
        const f32x4 x = x4[i * TS];
        s.x = fmaf(w4.x, x.x, d4.x * s.x);
        s.y = fmaf(w4.y, x.y, d4.y * s.y);
        s.z = fmaf(w4.z, x.z, d4.z * s.z);
        s.w = fmaf(w4.w, x.w, d4.w * s.w);
    }

    *(f32x4*)(carries + ((size_t)k * B + b) * C + c0) = s;
}

// ---------------------------------------------------------------------------
// Kernel 2: sequential combine of K chunk-carries per (b, channel-quad).
// In-place: carries[k] is read (offset s_k), then overwritten with the chunk's
// ENTRY state e_k.  e_0 = initial_state;  e_{k+1} = d^L * e_k + s_k.
// ---------------------------------------------------------------------------
__global__ __launch_bounds__(256) void ema_scan_carries(
    const float* __restrict__ initial_state,  // [B,C]
    const float* __restrict__ weights,        // [C]
    float* __restrict__ carries)              // [K,B,C] in/out
{
    const int tid = blockIdx.x * blockDim.x + threadIdx.x;  // 0 .. B*CG-1
    if (tid >= B * CG) return;
    const int cg = tid & (CG - 1);
    const int b  = tid >> 5;
    const int c0 = cg * CV;

    const f32x4 w4 = clamp01(*(const f32x4*)(weights + c0));
    f32x4 dL = {1.f - w4.x, 1.f - w4.y, 1.f - w4.z, 1.f - w4.w};
    // d^L with L = 64 = 2^6: six squarings.
#pragma unroll
    for (int j = 0; j < 6; ++j) {
        dL.x *= dL.x; dL.y *= dL.y; dL.z *= dL.z; dL.w *= dL.w;
    }

    f32x4 e = *(const f32x4*)(initial_state + (size_t)b * C + c0);

    for (int k = 0; k < K; ++k) {
        f32x4* slot = (f32x4*)(carries + ((size_t)k * B + b) * C + c0);
        const f32x4 s = *slot;   // chunk offset
        *slot = e;               // replace with chunk entry state
        e.x = fmaf(dL.x, e.x, s.x);
        e.y = fmaf(dL.y, e.y, s.y);
        e.z = fmaf(dL.z, e.z, s.z);
        e.w = fmaf(dL.w, e.w, s.w);
    }
}

// ---------------------------------------------------------------------------
// Kernel 3: finalize — rerun each chunk from its entry state.
//  * input re-read: NONTEMPORAL loads (last use; expected L2 hits from K1)
//  * output:        NONTEMPORAL stores (streamed once; don't thrash L2 and
//                   evict input lines other blocks still need)
// ---------------------------------------------------------------------------
__global__ __launch_bounds__(256) void ema_finalize(
    const float* __restrict__ inp,      // [B,T,C]
    const float* __restrict__ weights,  // [C]
    const float* __restrict__ carries,  // [K,B,C] = entry states
    float* __restrict__ out)            // [B,T,C]
{
    const int tid = blockIdx.x * blockDim.x + threadIdx.x;
    const int cg  = tid & (CG - 1);
    const int bk  = tid >> 5;
    const int k   = bk & (K - 1);
    const int b   = bk >> 7;
    const int c0  = cg * CV;

    const f32x4 w4 = clamp01(*(const f32x4*)(weights + c0));
    const f32x4 d4 = {1.f - w4.x, 1.f - w4.y, 1.f - w4.z, 1.f - w4.w};

    const size_t base = ((size_t)b * T + (size_t)k * L) * C + c0;
    const f32x4* __restrict__ x4 = (const f32x4*)(inp + base);
    f32x4* __restrict__ o4 = (f32x4*)(out + base);

    f32x4 s = *(const f32x4*)(carries + ((size_t)k * B + b) * C + c0);

    __builtin_prefetch(x4 + 16 * TS, 0, 0);
    __builtin_prefetch(x4 + 32 * TS, 0, 0);

#pragma unroll 16
    for (int i = 0; i < L; ++i) {
        if ((i & 15) == 0) __builtin_prefetch(x4 + (i + 48) * TS, 0, 0);
        const f32x4 x = __builtin_nontemporal_load(x4 + i * TS);
        s.x = fmaf(w4.x, x.x, d4.x * s.x);
        s.y = fmaf(w4.y, x.y, d4.y * s.y);
        s.z = fmaf(w4.z, x.z, d4.z * s.z);
        s.w = fmaf(w4.w, x.w, d4.w * s.w);
        __builtin_nontemporal_store(s, o4 + i * TS);
    }
}

// ---------------------------------------------------------------------------
extern "C" void kernel_launch(void* const* d_in, const int* in_sizes, int n_in,
                              void* d_out, int out_size, void* d_ws, size_t ws_size,
                              hipStream_t stream) {
    (void)in_sizes; (void)n_in; (void)out_size; (void)ws_size;
    const float* inputs        = (const float*)d_in[0];  // [B,T,C]
    const float* initial_state = (const float*)d_in[1];  // [B,C]
    const float* weights       = (const float*)d_in[2];  // [C]
    float* out = (float*)d_out;
    float* carries = (float*)d_ws;                       // K*B*C floats = 2 MB

    const int threads1 = B * K * CG;   // 131072
    dim3 blk(256);
    dim3 grd1(threads1 / 256);         // 512 blocks

    ema_partials<<<grd1, blk, 0, stream>>>(inputs, weights, carries);

    const int threads2 = B * CG;       // 1024
    dim3 grd2((threads2 + 255) / 256); // 4 blocks
    ema_scan_carries<<<grd2, blk, 0, stream>>>(initial_state, weights, carries);

    ema_finalize<<<grd1, blk, 0, stream>>>(inputs, weights, carries, out);
}